// MoleculeRegressor_83451214561990
// MI455X (gfx1250) — compile-verified
//
#include <hip/hip_runtime.h>
#include <hip/hip_bf16.h>

typedef __attribute__((ext_vector_type(2))) float v2f;
typedef __attribute__((ext_vector_type(8))) float v8f;
typedef int v4i_vs __attribute__((vector_size(4 * sizeof(int))));

#define N_NODES  50000
#define N_EDGES  800000
#define N_GRAPHS 512
#define F_IN     128
#define HID      256

// ---------------------------------------------------------------------------
// Degree: deg[dst]++ over edges, then dis = rsqrt(deg + 1)  (self loop)
// ---------------------------------------------------------------------------
__global__ void deg_accum_kernel(const int* __restrict__ dst, float* __restrict__ deg) {
    int e = blockIdx.x * blockDim.x + threadIdx.x;
    if (e < N_EDGES) atomicAdd(&deg[dst[e]], 1.0f);
}

__global__ void deg_finalize_kernel(float* __restrict__ deg) {
    int n = blockIdx.x * blockDim.x + threadIdx.x;
    if (n < N_NODES) deg[n] = rsqrtf(deg[n] + 1.0f);
}

// ---------------------------------------------------------------------------
// C[M,N] = A[M,K] @ B[K,N], fp32 via V_WMMA_F32_16X16X4_F32.
// Block = 256 threads = 8 waves; block tile = 16(M) x 128(N); one 16x16 tile
// per wave. The 16xK A stripe (contiguous, lda==K) is staged into LDS with a
// linear 16B copy — async global->LDS (B128, ASYNCcnt) when the toolchain
// exposes the builtin, plain copy otherwise. K, N are compile-time so all
// addressing is shifts (no integer division).
// ---------------------------------------------------------------------------
template <int K, int N>
__global__ __launch_bounds__(256)
void gemm_f32_wmma_kernel(const float* __restrict__ A, const float* __restrict__ B,
                          float* __restrict__ C) {
    __shared__ float As[16 * K];

    const int tid  = threadIdx.x;
    const int row0 = blockIdx.y * 16;

    // ---- stage A[row0:row0+16, 0:K]: one contiguous 16*K block ----
    constexpr int NV4 = 16 * K / 4;                 // 512 (K=128) or 1024 (K=256)
    const float4* A4  = (const float4*)(A + (size_t)row0 * K);
    float4*       As4 = (float4*)As;
#if __has_builtin(__builtin_amdgcn_global_load_async_to_lds_b128)
    #pragma unroll
    for (int i = tid; i < NV4; i += 256) {
        auto gsrc = (__attribute__((address_space(1))) v4i_vs*)(uintptr_t)(A4 + i);
        auto ldst = (__attribute__((address_space(3))) v4i_vs*)(unsigned)(uintptr_t)(As4 + i);
        __builtin_amdgcn_global_load_async_to_lds_b128(gsrc, ldst, 0, 0);
    }
  #if __has_builtin(__builtin_amdgcn_s_wait_asynccnt)
    __builtin_amdgcn_s_wait_asynccnt(0);
  #else
    asm volatile("s_wait_asynccnt 0x0" ::: "memory");
  #endif
#else
    #pragma unroll
    for (int i = tid; i < NV4; i += 256) As4[i] = A4[i];
#endif
    __syncthreads();

    const int lane = tid & 31;
    const int wave = tid >> 5;
    const int col0 = blockIdx.x * 128 + wave * 16;
    if (col0 >= N) return;                          // uniform per wave

    v8f acc = {0.f, 0.f, 0.f, 0.f, 0.f, 0.f, 0.f, 0.f};

    const int aRow = lane & 15;                     // A fragment: M = lane&15
    const int koff = (lane >> 4) << 1;              // lanes 16-31 hold K+2, K+3
    const int bcol = col0 + (lane & 15);

    #pragma unroll 8
    for (int k = 0; k < K; k += 4) {
        v2f a;
        a.x = As[aRow * K + k + koff];              // VGPR0: K = k   (or k+2)
        a.y = As[aRow * K + k + koff + 1];          // VGPR1: K = k+1 (or k+3)

        const float* bp = B + (size_t)(k + koff) * N + bcol;
        v2f b;
        b.x = bp[0];                                // row k+koff
        b.y = bp[N];                                // row k+koff+1

        acc = __builtin_amdgcn_wmma_f32_16x16x4_f32(
            /*neg_a=*/false, a, /*neg_b=*/false, b,
            /*c_mod=*/(short)0, acc, /*reuse_a=*/false, /*reuse_b=*/false);
    }

    // D layout: VGPR i -> rows {i, i+8}; lanes 16-31 take the +8 half
    const int crow = row0 + ((lane >> 4) << 3);
    float* cp = C + (size_t)crow * N + bcol;
    #pragma unroll
    for (int i = 0; i < 8; ++i) cp[(size_t)i * N] = acc[i];
}

// ---------------------------------------------------------------------------
// Edge aggregation: agg[dst] += hw[src] * dis[src]*dis[dst].
// 4 edges per block, 64 lanes per edge, float4 gathers (1KB/edge, coalesced).
// ---------------------------------------------------------------------------
__global__ __launch_bounds__(256)
void edge_agg_kernel(const float* __restrict__ hw, const int* __restrict__ src,
                     const int* __restrict__ dst, const float* __restrict__ dis,
                     float* __restrict__ agg) {
    int e  = blockIdx.x * 4 + (threadIdx.x >> 6);
    int c4 = (threadIdx.x & 63) << 2;
    int s = src[e];
    int d = dst[e];
    float norm = dis[s] * dis[d];
    const float4 v = *(const float4*)(hw + (size_t)s * HID + c4);
    float* ap = agg + (size_t)d * HID + c4;
    atomicAdd(ap + 0, v.x * norm);
    atomicAdd(ap + 1, v.y * norm);
    atomicAdd(ap + 2, v.z * norm);
    atomicAdd(ap + 3, v.w * norm);
}

// out = relu(agg + hw * dis^2 + bias)
__global__ void layer_finalize_kernel(const float* __restrict__ agg,
                                      const float* __restrict__ hw,
                                      const float* __restrict__ dis,
                                      const float* __restrict__ bias,
                                      float* __restrict__ out) {
    size_t idx = (size_t)blockIdx.x * blockDim.x + threadIdx.x;
    if (idx >= (size_t)N_NODES * HID) return;
    int n = (int)(idx >> 8);                        // HID == 256
    int c = (int)(idx & (HID - 1));
    float d = dis[n];
    float v = agg[idx] + hw[idx] * d * d + bias[c];
    out[idx] = fmaxf(v, 0.0f);
}

// ---------------------------------------------------------------------------
// Global mean pool
// ---------------------------------------------------------------------------
__global__ __launch_bounds__(256)
void pool_accum_kernel(const float* __restrict__ h, const int* __restrict__ batch,
                       float* __restrict__ sums, float* __restrict__ cnt) {
    int n = blockIdx.x;
    int c = threadIdx.x;
    int g = batch[n];
    atomicAdd(&sums[(size_t)g * HID + c], h[(size_t)n * HID + c]);
    if (c == 0) atomicAdd(&cnt[g], 1.0f);
}

__global__ void pool_div_kernel(float* __restrict__ sums, const float* __restrict__ cnt) {
    int idx = blockIdx.x * blockDim.x + threadIdx.x;
    if (idx < N_GRAPHS * HID) sums[idx] /= fmaxf(cnt[idx >> 8], 1.0f);
}

// z = relu(z + bm1)
__global__ void bias_relu_kernel(float* __restrict__ z, const float* __restrict__ b) {
    int idx = blockIdx.x * blockDim.x + threadIdx.x;
    if (idx < N_GRAPHS * HID) z[idx] = fmaxf(z[idx] + b[idx & (HID - 1)], 0.0f);
}

// out[g] = dot(z[g,:], Wm2[:,0]) + bm2
__global__ __launch_bounds__(256)
void mlp2_kernel(const float* __restrict__ z, const float* __restrict__ Wm2,
                 const float* __restrict__ bm2, float* __restrict__ out) {
    __shared__ float red[256];
    int g = blockIdx.x;
    int c = threadIdx.x;
    red[c] = z[(size_t)g * HID + c] * Wm2[c];
    __syncthreads();
    for (int s = 128; s > 0; s >>= 1) {
        if (c < s) red[c] += red[c + s];
        __syncthreads();
    }
    if (c == 0) out[g] = red[0] + bm2[0];
}

// ---------------------------------------------------------------------------
extern "C" void kernel_launch(void* const* d_in, const int* in_sizes, int n_in,
                              void* d_out, int out_size, void* d_ws, size_t ws_size,
                              hipStream_t stream) {
    const float* x     = (const float*)d_in[0];
    const int*   ei    = (const int*)d_in[1];      // [2, E]: src = ei[0:E], dst = ei[E:2E]
    const int*   batch = (const int*)d_in[2];
    const float* W0  = (const float*)d_in[3];  const float* b0  = (const float*)d_in[4];
    const float* W1  = (const float*)d_in[5];  const float* b1  = (const float*)d_in[6];
    const float* W2  = (const float*)d_in[7];  const float* b2  = (const float*)d_in[8];
    const float* Wm1 = (const float*)d_in[9];  const float* bm1 = (const float*)d_in[10];
    const float* Wm2 = (const float*)d_in[11]; const float* bm2 = (const float*)d_in[12];
    float* out = (float*)d_out;

    const int* src = ei;
    const int* dst = ei + N_EDGES;

    // workspace carve-out (256B aligned)
    char*  ws  = (char*)d_ws;
    size_t off = 0;
    auto carve = [&](size_t bytes) -> float* {
        float* p = (float*)(ws + off);
        off = (off + bytes + 255) & ~(size_t)255;
        return p;
    };
    float* dis  = carve((size_t)N_NODES * sizeof(float));
    float* hw   = carve((size_t)N_NODES * HID * sizeof(float));
    float* agg  = carve((size_t)N_NODES * HID * sizeof(float));
    float* hbuf = carve((size_t)N_NODES * HID * sizeof(float));
    float* sums = carve((size_t)N_GRAPHS * HID * sizeof(float));
    float* cnt  = carve((size_t)N_GRAPHS * sizeof(float));
    float* z    = carve((size_t)N_GRAPHS * HID * sizeof(float));

    // --- degrees ---
    (void)hipMemsetAsync(dis, 0, (size_t)N_NODES * sizeof(float), stream);
    deg_accum_kernel<<<(N_EDGES + 255) / 256, 256, 0, stream>>>(dst, dis);
    deg_finalize_kernel<<<(N_NODES + 255) / 256, 256, 0, stream>>>(dis);

    const size_t nh_bytes = (size_t)N_NODES * HID * sizeof(float);
    const dim3 gemm_grid_nodes(HID / 128, N_NODES / 16);
    const int  fin_blocks  = (int)(((size_t)N_NODES * HID + 255) / 256);
    const int  edge_blocks = N_EDGES / 4;

    // --- layer 0: x[50000,128] @ W0[128,256] ---
    gemm_f32_wmma_kernel<F_IN, HID><<<gemm_grid_nodes, 256, 0, stream>>>(x, W0, hw);
    (void)hipMemsetAsync(agg, 0, nh_bytes, stream);
    edge_agg_kernel<<<edge_blocks, 256, 0, stream>>>(hw, src, dst, dis, agg);
    layer_finalize_kernel<<<fin_blocks, 256, 0, stream>>>(agg, hw, dis, b0, hbuf);

    // --- layer 1 ---
    gemm_f32_wmma_kernel<HID, HID><<<gemm_grid_nodes, 256, 0, stream>>>(hbuf, W1, hw);
    (void)hipMemsetAsync(agg, 0, nh_bytes, stream);
    edge_agg_kernel<<<edge_blocks, 256, 0, stream>>>(hw, src, dst, dis, agg);
    layer_finalize_kernel<<<fin_blocks, 256, 0, stream>>>(agg, hw, dis, b1, hbuf);

    // --- layer 2 ---
    gemm_f32_wmma_kernel<HID, HID><<<gemm_grid_nodes, 256, 0, stream>>>(hbuf, W2, hw);
    (void)hipMemsetAsync(agg, 0, nh_bytes, stream);
    edge_agg_kernel<<<edge_blocks, 256, 0, stream>>>(hw, src, dst, dis, agg);
    layer_finalize_kernel<<<fin_blocks, 256, 0, stream>>>(agg, hw, dis, b2, hbuf);

    // --- global mean pool ---
    (void)hipMemsetAsync(sums, 0, (size_t)N_GRAPHS * HID * sizeof(float), stream);
    (void)hipMemsetAsync(cnt, 0, (size_t)N_GRAPHS * sizeof(float), stream);
    pool_accum_kernel<<<N_NODES, 256, 0, stream>>>(hbuf, batch, sums, cnt);
    pool_div_kernel<<<(N_GRAPHS * HID + 255) / 256, 256, 0, stream>>>(sums, cnt);

    // --- MLP: relu(pooled @ Wm1 + bm1) @ Wm2 + bm2 ---
    const dim3 gemm_grid_mlp(HID / 128, N_GRAPHS / 16);
    gemm_f32_wmma_kernel<HID, HID><<<gemm_grid_mlp, 256, 0, stream>>>(sums, Wm1, z);
    bias_relu_kernel<<<(N_GRAPHS * HID + 255) / 256, 256, 0, stream>>>(z, bm1);
    mlp2_kernel<<<N_GRAPHS, 256, 0, stream>>>(z, Wm2, bm2, out);
}